// ViolationInformedLossAccelerated_77051713290883
// MI455X (gfx1250) — compile-verified
//
#include <hip/hip_runtime.h>
#include <math.h>

// ---------------------------------------------------------------------------
// ViolationInformedLoss for MI455X (gfx1250, wave32, WMMA + async->LDS)
//   out[0] = total_loss, out[1] = loss_logmse, out[2] = loss_violation
// ---------------------------------------------------------------------------

typedef __attribute__((ext_vector_type(2))) float v2f;
typedef __attribute__((ext_vector_type(8))) float v8f;
typedef __attribute__((ext_vector_type(4))) int   v4i;

#define EPSF        1e-9f
#define LAMBDA_V    0.5f
#define BATCH       4096
#define SROWS       64      // NUM_SAMPLES
#define KD          64      // K
#define ND          128     // N
#define MTOP        8       // M
#define STR         68      // padded LDS row stride (floats): 272 B per row ->
                            // strided-by-row A reads hit distinct banks, and
                            // rows stay 16-B aligned for b128 LDS writes

#if __has_builtin(__builtin_amdgcn_global_load_async_to_lds_b128) && \
    __has_builtin(__builtin_amdgcn_s_wait_asynccnt)
#define VI_HAS_ASYNC_LDS 1
typedef v4i __attribute__((address_space(1)))* vi_gptr_t;   // global, 64-bit
typedef v4i __attribute__((address_space(3)))* vi_lptr_t;   // LDS, 32-bit
#else
#define VI_HAS_ASYNC_LDS 0
#endif

// ---------------- threefry2x32-20 (JAX PRNG), key = PRNGKey(42) = {0,42} ----
__device__ __forceinline__ unsigned rotl32(unsigned x, int r) {
  return (x << r) | (x >> (32 - r));
}

__device__ __forceinline__ void threefry2x32(unsigned x0, unsigned x1,
                                             unsigned& o0, unsigned& o1) {
  const unsigned ks0 = 0u;
  const unsigned ks1 = 42u;
  const unsigned ks2 = ks0 ^ ks1 ^ 0x1BD11BDAu;
  x0 += ks0; x1 += ks1;
#define TF_R4(ra, rb, rc, rd)                                  \
  x0 += x1; x1 = rotl32(x1, ra); x1 ^= x0;                     \
  x0 += x1; x1 = rotl32(x1, rb); x1 ^= x0;                     \
  x0 += x1; x1 = rotl32(x1, rc); x1 ^= x0;                     \
  x0 += x1; x1 = rotl32(x1, rd); x1 ^= x0;
  TF_R4(13, 15, 26, 6)   x0 += ks1; x1 += ks2 + 1u;
  TF_R4(17, 29, 16, 24)  x0 += ks2; x1 += ks0 + 2u;
  TF_R4(13, 15, 26, 6)   x0 += ks0; x1 += ks1 + 3u;
  TF_R4(17, 29, 16, 24)  x0 += ks1; x1 += ks2 + 4u;
  TF_R4(13, 15, 26, 6)   x0 += ks2; x1 += ks0 + 5u;
#undef TF_R4
  o0 = x0; o1 = x1;
}

// bits -> uniform(-0.99999994, 1) -> sqrt(2)*erfinv(u)  (Giles single-prec)
__device__ __forceinline__ float bits_to_normal(unsigned bits) {
  unsigned fb = (bits >> 9) | 0x3f800000u;       // [1,2)
  float f01 = __uint_as_float(fb) - 1.0f;        // [0,1)
  const float lo = -0.99999994f, hi = 1.0f;
  float u = f01 * (hi - lo) + lo;
  float w = -logf((1.0f - u) * (1.0f + u));
  float p;
  if (w < 5.0f) {
    w -= 2.5f;
    p = 2.81022636e-08f;
    p = 3.43273939e-07f + p * w;
    p = -3.5233877e-06f + p * w;
    p = -4.39150654e-06f + p * w;
    p = 0.00021858087f + p * w;
    p = -0.00125372503f + p * w;
    p = -0.00417768164f + p * w;
    p = 0.246640727f + p * w;
    p = 1.50140941f + p * w;
  } else {
    w = sqrtf(w) - 3.0f;
    p = -0.000200214257f;
    p = 0.000100950558f + p * w;
    p = 0.00134934322f + p * w;
    p = -0.00367342844f + p * w;
    p = 0.00573950773f + p * w;
    p = -0.0076224613f + p * w;
    p = 0.00943887047f + p * w;
    p = 1.00167406f + p * w;
    p = 2.83297682f + p * w;
  }
  return 1.4142135623730951f * p * u;
}

// ---------------- kernel 0: zero accumulators -------------------------------
__global__ void vi_init_kernel(float* __restrict__ acc) {
  acc[0] = 0.0f;  // sum of squared log2 errors
  acc[1] = 0.0f;  // sum of penalties
}

// ---------------- kernel 1: logmse reduction --------------------------------
__global__ __launch_bounds__(256) void vi_logmse_kernel(
    const float* __restrict__ y_pred, const float* __restrict__ y_true,
    float* __restrict__ acc) {
  __shared__ float red[256];
  int i = blockIdx.x * 256 + threadIdx.x;
  float d = 0.0f;
  if (i < BATCH) {
    float lt = log2f(fmaxf(y_true[i], EPSF));
    float lp = log2f(fmaxf(y_pred[i], EPSF));
    float e = lt - lp;
    d = e * e;
  }
  red[threadIdx.x] = d;
  __syncthreads();
  for (int s = 128; s > 0; s >>= 1) {
    if ((int)threadIdx.x < s) red[threadIdx.x] += red[threadIdx.x + s];
    __syncthreads();
  }
  if (threadIdx.x == 0) atomicAdd(&acc[0], red[0]);
}

// ---------------- kernel 2: per-batch GEMM (WMMA) + top-k penalty -----------
// One workgroup (256 thr = 8 wave32) per batch element.
//   C = [X | X @ P]  (64 x 128); only the right half needs the GEMM.
__global__ __launch_bounds__(256) void vi_gemm_topk_kernel(
    const float* __restrict__ y_pred, const float* __restrict__ P_padded,
    float* __restrict__ acc) {
  __shared__ float Xl[SROWS * STR];
  __shared__ float Pl[KD * STR];
  __shared__ float Rl[SROWS * STR];
  __shared__ float hm_l[SROWS];

  const int b   = blockIdx.x;
  const int tid = threadIdx.x;

  const float* Pg = P_padded + (size_t)b * (KD * (ND - KD));

#if VI_HAS_ASYNC_LDS
  // ---- async DMA P (64x64 f32) global->LDS, b128 per lane; the transfer
  //      overlaps the VALU-heavy X generation below (tracked by ASYNCcnt).
#pragma unroll
  for (int i = 0; i < 4; ++i) {
    int idx = tid + i * 256;     // 1024 16-B chunks
    int row = idx >> 4;          // 16 chunks per 64-float row
    int c4  = idx & 15;
    vi_gptr_t gsrc = (vi_gptr_t)(uintptr_t)(Pg + idx * 4);
    vi_lptr_t ldst =
        (vi_lptr_t)(uint32_t)(uintptr_t)&Pl[row * STR + c4 * 4];
    __builtin_amdgcn_global_load_async_to_lds_b128(gsrc, ldst, 0, 0);
  }
#else
  // ---- fallback: synchronous float4 load + ds_store ----
  const float4* Pg4 = (const float4*)Pg;
#pragma unroll
  for (int i = 0; i < 4; ++i) {
    int idx = tid + i * 256;     // 1024 float4 total
    int row = idx >> 4;          // 16 float4 per row of 64 floats
    int c4  = idx & 15;
    float4 v = Pg4[idx];
    float* dst = &Pl[row * STR + c4 * 4];
    dst[0] = v.x; dst[1] = v.y; dst[2] = v.z; dst[3] = v.w;
  }
#endif

  // ---- generate X (64x64) = jax.random.normal(key(42), (B,64,64)) ----
  const unsigned half_n = (unsigned)(BATCH * SROWS * KD) / 2u;  // 8388608
#pragma unroll
  for (int i = 0; i < 16; ++i) {
    int idx = tid + i * 256;                       // 0..4095
    unsigned g = (unsigned)b * 4096u + (unsigned)idx;
    unsigned c0, c1, o0, o1;
    bool first = (g < half_n);
    if (first) { c0 = g; c1 = g + half_n; } else { c0 = g - half_n; c1 = g; }
    threefry2x32(c0, c1, o0, o1);
    unsigned bits = first ? o0 : o1;
    int row = idx >> 6, col = idx & 63;
    Xl[row * STR + col] = bits_to_normal(bits);
  }

#if VI_HAS_ASYNC_LDS
  __builtin_amdgcn_s_wait_asynccnt(0);  // P tile resident in LDS
#endif
  __syncthreads();

  // ---- R = X @ P via V_WMMA_F32_16X16X4_F32 ----
  // 16 output tiles (4x4 of 16x16); 8 waves -> 2 tiles each; 16 k-steps of 4.
  const int wave   = tid >> 5;
  const int lane   = tid & 31;
  const int half   = lane >> 4;   // 0: K{0,1} / M0-7 rows of D ; 1: K{2,3} / M8-15
  const int lane16 = lane & 15;

#pragma unroll
  for (int t = 2 * wave; t < 2 * wave + 2; ++t) {
    const int mt = (t >> 2) * 16;   // tile row base in R
    const int nt = (t & 3) * 16;    // tile col base in R
    v8f accv = {};
#pragma unroll
    for (int kk = 0; kk < KD; kk += 4) {
      const int ko = kk + half * 2;
      v2f a, bb;
      a.x  = Xl[(mt + lane16) * STR + ko];
      a.y  = Xl[(mt + lane16) * STR + ko + 1];
      bb.x = Pl[ko * STR + nt + lane16];
      bb.y = Pl[(ko + 1) * STR + nt + lane16];
      accv = __builtin_amdgcn_wmma_f32_16x16x4_f32(
          /*neg_a=*/false, a, /*neg_b=*/false, bb,
          /*c_mod=*/(short)0, accv, /*reuse_a=*/false, /*reuse_b=*/false);
    }
    // D layout: VGPR r holds M = r + 8*half, N = lane16
#pragma unroll
    for (int r = 0; r < 8; ++r) {
      Rl[(mt + r + half * 8) * STR + nt + lane16] = accv[r];
    }
  }
  __syncthreads();

  // ---- per-row top-(M+1) over 128 mags: [ |X row| , |R row| ] ----
  if (tid < SROWS) {
    float top[MTOP + 1];
#pragma unroll
    for (int r = 0; r <= MTOP; ++r) top[r] = -1.0f;
    const float* xr = &Xl[tid * STR];
    const float* rr = &Rl[tid * STR];
    for (int j = 0; j < 2 * KD; ++j) {
      float v = fabsf(j < KD ? xr[j] : rr[j - KD]);
      if (v > top[MTOP]) {
        top[MTOP] = v;
#pragma unroll
        for (int r = MTOP; r > 0; --r) {
          if (top[r] > top[r - 1]) {
            float tmp = top[r]; top[r] = top[r - 1]; top[r - 1] = tmp;
          }
        }
      }
    }
    hm_l[tid] = top[0] / (top[MTOP] + EPSF);
  }
  __syncthreads();

  if (tid == 0) {
    float mx = hm_l[0];
    for (int r = 1; r < SROWS; ++r) mx = fmaxf(mx, hm_l[r]);
    float pen = fmaxf(mx - y_pred[b], 0.0f);
    atomicAdd(&acc[1], pen);
  }
}

// ---------------- kernel 3: finalize ----------------------------------------
__global__ void vi_finalize_kernel(const float* __restrict__ acc,
                                   float* __restrict__ out) {
  float lm  = acc[0] / (float)BATCH;
  float pen = acc[1] / (float)BATCH;
  out[0] = lm + LAMBDA_V * pen;
  out[1] = lm;
  out[2] = pen;
}

// ---------------------------------------------------------------------------
extern "C" void kernel_launch(void* const* d_in, const int* in_sizes, int n_in,
                              void* d_out, int out_size, void* d_ws,
                              size_t ws_size, hipStream_t stream) {
  const float* y_pred   = (const float*)d_in[0];  // (B,1)
  const float* y_true   = (const float*)d_in[1];  // (B,1)
  const float* P_padded = (const float*)d_in[2];  // (B,64,64)
  // d_in[3] = params (int64 [N,K,M] tiled) -> compile-time constants here.
  (void)in_sizes; (void)n_in; (void)out_size; (void)ws_size;

  float* out = (float*)d_out;
  float* acc = (float*)d_ws;  // [0]=logmse sum, [1]=penalty sum

  vi_init_kernel<<<1, 1, 0, stream>>>(acc);
  vi_logmse_kernel<<<BATCH / 256, 256, 0, stream>>>(y_pred, y_true, acc);
  vi_gemm_topk_kernel<<<BATCH, 256, 0, stream>>>(y_pred, P_padded, acc);
  vi_finalize_kernel<<<1, 1, 0, stream>>>(acc, out);
}